// BalanceL1Loss_2731599200959
// MI455X (gfx1250) — compile-verified
//
#include <hip/hip_runtime.h>
#include <hip/hip_bf16.h>

typedef __attribute__((ext_vector_type(2))) float v2f;
typedef __attribute__((ext_vector_type(8))) float v8f;

#define BINS       8000      // bits>>18 : exp(8) + top-5 mantissa bits, clamped
#define NBLK       512
#define NTHR       256
#define NWAVES     (NTHR / 32)
#define NEG_RATIO  3.0f

// ---------------- workspace layout (in d_ws) ----------------
// float    gacc[4]   : pos_num, neg_cnt, pos_loss_sum, total_loss_sum
// unsigned gcnt[BINS]: global histogram counts of negative losses
// float    gsum[BINS]: global histogram value-sums
// total: 16 + 8000*4 + 8000*4 = 64016 bytes

__global__ void bl1_init(float* gacc, unsigned* gcnt, float* gsum) {
    int i = blockIdx.x * blockDim.x + threadIdx.x;
    if (i < 4) gacc[i] = 0.0f;
    if (i < BINS) { gcnt[i] = 0u; gsum[i] = 0.0f; }
}

__device__ __forceinline__ float v8sum(v8f v) {
    return ((v[0] + v[1]) + (v[2] + v[3])) + ((v[4] + v[5]) + (v[6] + v[7]));
}

__device__ __forceinline__ v8f wmma_acc(float x0, float x1, v2f ones, v8f c) {
    v2f a; a[0] = x0; a[1] = x1;
    // D = A(16x4) * ones(4x16) + C : every D column accumulates the row-sums of A.
    return __builtin_amdgcn_wmma_f32_16x16x4_f32(
        /*neg_a=*/false, a, /*neg_b=*/false, ones,
        /*c_mod=*/(short)0, c, /*reuse_a=*/false, /*reuse_b=*/false);
}

__global__ __launch_bounds__(NTHR)
void bl1_pass1(const float* __restrict__ pred, const float* __restrict__ gt,
               const float* __restrict__ mask, long long n,
               float* gacc, unsigned* gcnt, float* gsum) {
    __shared__ unsigned hcnt[BINS];
    __shared__ float    hsum[BINS];
    __shared__ float    wred[NWAVES * 4];

    for (int b = threadIdx.x; b < BINS; b += NTHR) { hcnt[b] = 0u; hsum[b] = 0.0f; }
    __syncthreads();

    const long long n4     = n >> 2;
    const long long stride = (long long)gridDim.x * NTHR;
    const long long tid    = (long long)blockIdx.x * NTHR + threadIdx.x;
    const long long iters  = (n4 + stride - 1) / stride;   // uniform trip count

    const float4* p4 = (const float4*)pred;
    const float4* g4 = (const float4*)gt;
    const float4* m4 = (const float4*)mask;

    v2f ones; ones[0] = 1.0f; ones[1] = 1.0f;
    v8f accPos = {}, accNeg = {}, accPL = {}, accTL = {};
    float tq0 = 0.f, tq1 = 0.f, tq2 = 0.f, tq3 = 0.f;   // scalar tail accumulators

    for (long long it = 0; it < iters; ++it) {
        long long i = it * stride + tid;
        float4 p = make_float4(0.f, 0.f, 0.f, 0.f);
        float4 g = p, m = p;
        if (i < n4) {
            p = p4[i]; g = g4[i]; m = m4[i];
            __builtin_prefetch(&p4[i + stride], 0, 0);   // global_prefetch_b8
            __builtin_prefetch(&g4[i + stride], 0, 0);
            __builtin_prefetch(&m4[i + stride], 0, 0);
        }
        float pe[4] = {p.x, p.y, p.z, p.w};
        float ge[4] = {g.x, g.y, g.z, g.w};
        float me[4] = {m.x, m.y, m.z, m.w};
        float pos[4], neg[4], pl[4], tl[4];
#pragma unroll
        for (int j = 0; j < 4; ++j) {
            float bin  = (ge[j] > 0.0f) ? 1.0f : 0.0f;
            float loss = fabsf(pe[j] - ge[j]);
            pos[j] = bin * me[j];
            neg[j] = (1.0f - bin) * me[j];
            pl[j]  = pos[j] * loss;
            tl[j]  = loss;
            if (neg[j] > 0.0f) {
                unsigned ix = __float_as_uint(loss) >> 18;
                if (ix > BINS - 1) ix = BINS - 1;
                atomicAdd(&hcnt[ix], 1u);
                atomicAdd(&hsum[ix], loss);
            }
        }
        // XDL-pipe accumulation: 8 wmmas consume 4 elements/lane for 4 quantities.
        accPos = wmma_acc(pos[0], pos[1], ones, accPos);
        accPos = wmma_acc(pos[2], pos[3], ones, accPos);
        accNeg = wmma_acc(neg[0], neg[1], ones, accNeg);
        accNeg = wmma_acc(neg[2], neg[3], ones, accNeg);
        accPL  = wmma_acc(pl[0],  pl[1],  ones, accPL);
        accPL  = wmma_acc(pl[2],  pl[3],  ones, accPL);
        accTL  = wmma_acc(tl[0],  tl[1],  ones, accTL);
        accTL  = wmma_acc(tl[2],  tl[3],  ones, accTL);
    }

    // scalar tail (n % 4), handled by block 0 only
    long long rem = n - (n4 << 2);
    if (blockIdx.x == 0 && (long long)threadIdx.x < rem) {
        long long e = (n4 << 2) + threadIdx.x;
        float pe2 = pred[e], ge2 = gt[e], me2 = mask[e];
        float bin  = (ge2 > 0.0f) ? 1.0f : 0.0f;
        float loss = fabsf(pe2 - ge2);
        float po = bin * me2, ne = (1.0f - bin) * me2;
        tq0 += po; tq1 += ne; tq2 += po * loss; tq3 += loss;
        if (ne > 0.0f) {
            unsigned ix = __float_as_uint(loss) >> 18;
            if (ix > BINS - 1) ix = BINS - 1;
            atomicAdd(&hcnt[ix], 1u);
            atomicAdd(&hsum[ix], loss);
        }
    }

    // every D column is identical -> divide the full-tile sum by 16
    float q[4];
    q[0] = v8sum(accPos) * 0.0625f + tq0;
    q[1] = v8sum(accNeg) * 0.0625f + tq1;
    q[2] = v8sum(accPL)  * 0.0625f + tq2;
    q[3] = v8sum(accTL)  * 0.0625f + tq3;

    // wave32 shuffle reduction, then per-wave leaders -> thread 0
    for (int off = 16; off > 0; off >>= 1)
#pragma unroll
        for (int i = 0; i < 4; ++i) q[i] += __shfl_down(q[i], off, 32);
    int lane = threadIdx.x & 31, wid = threadIdx.x >> 5;
    if (lane == 0)
        for (int i = 0; i < 4; ++i) wred[wid * 4 + i] = q[i];
    __syncthreads();
    if (threadIdx.x == 0) {
        float s[4] = {0.f, 0.f, 0.f, 0.f};
        for (int w = 0; w < NWAVES; ++w)
            for (int i = 0; i < 4; ++i) s[i] += wred[w * 4 + i];
        for (int i = 0; i < 4; ++i) atomicAdd(&gacc[i], s[i]);
    }
    __syncthreads();

    // flush block-local histogram once
    for (int b = threadIdx.x; b < BINS; b += NTHR) {
        unsigned c = hcnt[b];
        if (c) { atomicAdd(&gcnt[b], c); atomicAdd(&gsum[b], hsum[b]); }
    }
}

__global__ __launch_bounds__(256)
void bl1_finalize(const float* gacc, const unsigned* gcnt, const float* gsum,
                  float* out, long long n) {
    const int CH = 32, NCH = BINS / CH;   // 250 chunks
    __shared__ unsigned long long ccnt[256];
    __shared__ float csum[256];
    int t = threadIdx.x;
    unsigned long long c = 0; float s = 0.0f;
    if (t < NCH)
        for (int b = t * CH; b < t * CH + CH; ++b) { c += gcnt[b]; s += gsum[b]; }
    ccnt[t] = c; csum[t] = s;
    __syncthreads();

    if (t == 0) {
        float pos_num  = gacc[0];
        float neg_cnt  = gacc[1];
        float pos_loss = gacc[2];
        float tot_loss = gacc[3];
        float neg_num  = fminf(neg_cnt, pos_num * NEG_RATIO);
        double k = floorf(neg_num);

        double cum = 0.0, sum_above = 0.0, neg_loss = 0.0;
        bool found = false;
        for (int ch = NCH - 1; ch >= 0 && !found; --ch) {
            if (cum + (double)ccnt[ch] >= k) {
                for (int b = ch * CH + CH - 1; b >= ch * CH; --b) {
                    unsigned cb = gcnt[b];
                    if (cum + (double)cb >= k) {
                        double take = k - cum;
                        double avg  = cb ? (double)gsum[b] / (double)cb : 0.0;
                        neg_loss = sum_above + take * avg;
                        found = true; break;
                    }
                    cum += (double)cb; sum_above += (double)gsum[b];
                }
            } else {
                cum += (double)ccnt[ch]; sum_above += (double)csum[ch];
            }
        }
        if (!found) neg_loss = sum_above;   // k >= all negatives: keep everything

        float balance = (float)(((double)pos_loss + neg_loss) /
                                ((double)pos_num + (double)neg_num + 1e-6));
        out[0] = (pos_num == 0.0f) ? (tot_loss / (float)n) : balance;
    }
}

extern "C" void kernel_launch(void* const* d_in, const int* in_sizes, int n_in,
                              void* d_out, int out_size, void* d_ws, size_t ws_size,
                              hipStream_t stream) {
    const float* pred = (const float*)d_in[0];
    const float* gt   = (const float*)d_in[1];
    const float* mask = (const float*)d_in[2];
    float* out = (float*)d_out;
    long long n = (long long)in_sizes[0];

    float*    gacc = (float*)d_ws;
    unsigned* gcnt = (unsigned*)(gacc + 4);
    float*    gsum = (float*)(gcnt + BINS);

    bl1_init<<<(BINS + 255) / 256, 256, 0, stream>>>(gacc, gcnt, gsum);
    bl1_pass1<<<NBLK, NTHR, 0, stream>>>(pred, gt, mask, n, gacc, gcnt, gsum);
    bl1_finalize<<<1, 256, 0, stream>>>(gacc, gcnt, gsum, out, n);
}